// EVI_Conv2D_19353122636472
// MI455X (gfx1250) — compile-verified
//
#include <hip/hip_runtime.h>
#include <hip/hip_bf16.h>
#include <cstddef>

typedef __attribute__((ext_vector_type(2))) float v2f;
typedef __attribute__((ext_vector_type(8))) float v8f;

#define B_   16
#define CIN  16
#define H_   24
#define KS   5
#define COUT 32
#define HO   20
#define L_   400   // HO*HO
#define D_   400   // CIN*KS*KS
#define HW   576   // H_*H_

// softplus of sigma_w -> sp[COUT]
__global__ void sp_kernel(const float* __restrict__ sigma_w, float* __restrict__ sp) {
    int t = threadIdx.x;
    if (t < COUT) sp[t] = log1pf(expf(sigma_w[t]));
}

// diagonal of sigma -> sigma_img [B,CIN,H,H]; sigma is 340MB read-once -> NT loads
__global__ __launch_bounds__(256)
void diag_kernel(const float* __restrict__ sigma, float* __restrict__ sig_img) {
    int t = blockIdx.x * 256 + threadIdx.x;
    if (t >= B_ * CIN * HW) return;
    int i  = t % HW;
    int bc = t / HW;
    sig_img[t] = __builtin_nontemporal_load(&sigma[((size_t)bc * HW + i) * HW + i]);
}

// im2col of mu -> xp [B, L, D] row-major (torch Unfold ordering: c, kh, kw)
__global__ __launch_bounds__(128)
void xp_kernel(const float* __restrict__ mu, float* __restrict__ xp) {
    const int l = blockIdx.x;           // 0..L-1
    const int b = blockIdx.y;           // 0..B-1
    const int y = l / HO, x = l - y * HO;
    float* xrow = xp + (size_t)(b * L_ + l) * D_;
    for (int d = threadIdx.x; d < D_; d += 128) {
        int c   = d / (KS * KS);
        int rem = d - c * (KS * KS);
        int kh  = rem / KS, kw = rem - kh * KS;
        xrow[d] = mu[((size_t)(b * CIN + c) * H_ + (y + kh)) * H_ + (x + kw)];
    }
}

// one wave per (b,l); lane = cout. Computes mu_z and fused diag term:
// dterm[b,co,l] = sum_d (w[co,d]^2 + sp[co]) * sigma_patch[b,d,l]
//              = mu_w^2-term + sp*trace
__global__ __launch_bounds__(32)
void muzdt_kernel(const float* __restrict__ xp, const float* __restrict__ sig_img,
                  const float* __restrict__ w,  const float* __restrict__ sp,
                  float* __restrict__ mu_z, float* __restrict__ dterm) {
    const int bl = blockIdx.x;          // b*L + l
    const int b  = bl / L_;
    const int l  = bl - b * L_;
    const int y  = l / HO, x = l - y * HO;
    const int co = threadIdx.x;         // 0..31
    const float spc = sp[co];
    const float* xrow = xp + (size_t)bl * D_;
    const float* wrow = w + co * D_;
    float accm = 0.f, accd = 0.f;
    int d = 0;
    for (int c = 0; c < CIN; ++c) {
        const float* sb = sig_img + (size_t)(b * CIN + c) * HW + y * H_ + x;
        for (int kh = 0; kh < KS; ++kh) {
            const float* srow = sb + kh * H_;
            #pragma unroll
            for (int kw = 0; kw < KS; ++kw) {
                float wv = wrow[d];
                accm += wv * xrow[d];           // conv (mean path)
                float sv = srow[kw];            // broadcast across lanes
                accd += (wv * wv + spc) * sv;
                ++d;
            }
        }
    }
    mu_z [(size_t)(b * COUT + co) * L_ + l] = accm;
    dterm[(size_t)(b * COUT + co) * L_ + l] = accd;
}

// Gram G = X X^T per batch via V_WMMA_F32_16X16X4_F32, fused scale-by-sp and
// diagonal add, fanned out across all 32 couts from registers with NT stores.
__global__ __launch_bounds__(128)
void gram_kernel(const float* __restrict__ xp, const float* __restrict__ sp,
                 const float* __restrict__ dterm, float* __restrict__ sigma_z) {
    __shared__ float a_lds[D_ * 16];   // k-major: a_lds[k*16 + r] = X[l0+r][k]
    __shared__ float sp_lds[COUT];
    const int l0t = blockIdx.x;        // tile row index, 0..24
    const int b   = blockIdx.y;
    const int tid = threadIdx.x;
    if (tid < COUT) sp_lds[tid] = sp[tid];

    const float* xb = xp + (size_t)b * L_ * D_;
    // stage the 16 A rows (coalesced global reads, k-major LDS writes)
    for (int idx = tid; idx < 16 * D_; idx += 128) {
        int r = idx / D_;
        int k = idx - r * D_;
        a_lds[k * 16 + r] = xb[(size_t)(l0t * 16 + r) * D_ + k];
    }
    __syncthreads();

    const int lane = tid & 31;
    const int wid  = tid >> 5;
    const int lr   = lane & 15;
    const int koff = (lane < 16) ? 0 : 2;   // A/B 16x4 f32 layout: hi-half lanes hold K=2,3

    const float* dterm_b = dterm + (size_t)b * COUT * L_;

    for (int m0t = wid; m0t < L_ / 16; m0t += 4) {
        const float* brow = xb + (size_t)(m0t * 16 + lr) * D_ + koff; // 8B-aligned
        v8f acc = {};
        #pragma unroll 4
        for (int k = 0; k < D_; k += 4) {
            v2f a;
            a.x = a_lds[(k + koff) * 16 + lr];
            a.y = a_lds[(k + koff + 1) * 16 + lr];
            v2f bv = *(const v2f*)(brow + k);
            acc = __builtin_amdgcn_wmma_f32_16x16x4_f32(
                false, a, false, bv, (short)0, acc, false, false);
        }
        const int  col      = m0t * 16 + lr;
        const int  rbase    = l0t * 16 + ((lane < 16) ? 0 : 8);
        const int  rdiag    = col - rbase;          // in [0,8) for the diag lane only
        const bool diagtile = (m0t * 16 == l0t * 16); // wave-uniform

        for (int co = 0; co < COUT; ++co) {
            float s  = sp_lds[co];
            float dl = 0.f;
            if (diagtile)                           // uniform branch, coalesced 64B load
                dl = dterm_b[co * L_ + col];        // dterm[row]==dterm[col] iff row==col
            float* outp = sigma_z + (size_t)(b * COUT + co) * L_ * L_
                        + (size_t)rbase * L_ + col;
            #pragma unroll
            for (int r = 0; r < 8; ++r) {
                float v = s * acc[r] + ((r == rdiag) ? dl : 0.f);  // branch-free cndmask
                __builtin_nontemporal_store(v, outp + (size_t)r * L_);
            }
        }
    }
}

extern "C" void kernel_launch(void* const* d_in, const int* in_sizes, int n_in,
                              void* d_out, int out_size, void* d_ws, size_t ws_size,
                              hipStream_t stream) {
    const float* mu      = (const float*)d_in[0];
    const float* sigma   = (const float*)d_in[1];
    const float* weight  = (const float*)d_in[2];
    const float* sigma_w = (const float*)d_in[3];

    float* out     = (float*)d_out;
    float* mu_z    = out;                                  // B*COUT*L
    float* sigma_z = out + (size_t)B_ * COUT * L_;         // B*COUT*L*L

    float* xp      = (float*)d_ws;                         // B*L*D      (10.24 MB)
    float* sig_img = xp      + (size_t)B_ * L_ * D_;       // B*CIN*HW   (0.59 MB)
    float* dterm   = sig_img + (size_t)B_ * CIN * HW;      // B*COUT*L   (0.82 MB)
    float* spv     = dterm   + (size_t)B_ * COUT * L_;     // COUT

    sp_kernel   <<<1, 32, 0, stream>>>(sigma_w, spv);
    diag_kernel <<<(B_ * CIN * HW + 255) / 256, 256, 0, stream>>>(sigma, sig_img);
    xp_kernel   <<<dim3(L_, B_), 128, 0, stream>>>(mu, xp);
    muzdt_kernel<<<B_ * L_, 32, 0, stream>>>(xp, sig_img, weight, spv, mu_z, dterm);
    gram_kernel <<<dim3(L_ / 16, B_), 128, 0, stream>>>(xp, spv, dterm, sigma_z);
}